// MilpGNN_46179488367114
// MI455X (gfx1250) — compile-verified
//
#include <hip/hip_runtime.h>

typedef __attribute__((ext_vector_type(2))) float v2f;
typedef __attribute__((ext_vector_type(8))) float v8f;

#define HD 64

// Native hardware FP atomic add on global memory (global_atomic_add_f32),
// bypassing the CAS-loop fallback hipcc can emit for plain atomicAdd(float*).
__device__ __forceinline__ void gAddF(float* p, float v) { unsafeAtomicAdd(p, v); }

// ---------------------------------------------------------------- degrees
__global__ __launch_bounds__(256) void k_count_deg(const int* __restrict__ src,
                                                   const int* __restrict__ dst,
                                                   int E, float* __restrict__ degC,
                                                   float* __restrict__ degV) {
  int e = blockIdx.x * 256 + threadIdx.x;
  if (e < E) {
    gAddF(&degC[src[e]], 1.0f);
    gAddF(&degV[dst[e]], 1.0f);
  }
}

__global__ __launch_bounds__(256) void k_inv_clamp(float* __restrict__ d, int n) {
  int i = blockIdx.x * 256 + threadIdx.x;
  if (i < n) d[i] = 1.0f / fmaxf(d[i], 1.0f);
}

// ---------------------------------------------------------------- BN stats
// Small-C (1 or 9) stats via LDS atomics; layer-0 only.
__global__ __launch_bounds__(256) void k_bn_stats_small(const float* __restrict__ x,
                                                        int total, int C,
                                                        float* __restrict__ stats) {
  __shared__ float ls[32];
  if (threadIdx.x < 32) ls[threadIdx.x] = 0.0f;
  __syncthreads();
  for (int i = blockIdx.x * 256 + threadIdx.x; i < total; i += gridDim.x * 256) {
    float v = x[i];
    int c = i % C;
    atomicAdd(&ls[c], v);
    atomicAdd(&ls[C + c], v * v);
  }
  __syncthreads();
  if (threadIdx.x < 2 * C) gAddF(&stats[threadIdx.x], ls[threadIdx.x]);
}

// C==64: each thread owns one fixed column -> register accumulation, coalesced.
__global__ __launch_bounds__(256) void k_bn_stats64(const float* __restrict__ x, int n,
                                                    float* __restrict__ stats) {
  __shared__ float ss[128];
  if (threadIdx.x < 128) ss[threadIdx.x] = 0.0f;
  __syncthreads();
  int col = threadIdx.x & 63;
  int rsub = threadIdx.x >> 6;  // 0..3
  float s = 0.0f, q = 0.0f;
  for (int r = blockIdx.x * 4 + rsub; r < n; r += gridDim.x * 4) {
    float v = x[(size_t)r * HD + col];
    s += v;
    q += v * v;
  }
  atomicAdd(&ss[col], s);
  atomicAdd(&ss[64 + col], q);
  __syncthreads();
  if (threadIdx.x < 128) gAddF(&stats[threadIdx.x], ss[threadIdx.x]);
}

__global__ void k_bn_finalize(const float* __restrict__ stats, int C, float invN,
                              const float* __restrict__ g, const float* __restrict__ b,
                              float* __restrict__ sc, float* __restrict__ sh) {
  int i = threadIdx.x;
  if (i < C) {
    float m = stats[i] * invN;
    float v = stats[C + i] * invN - m * m;
    float s = g[i] * rsqrtf(v + 1e-5f);
    sc[i] = s;
    sh[i] = b[i] - m * s;
  }
}

// ---------------------------------------------------------------- layer-0 aggregation
__global__ __launch_bounds__(256) void k_agg0_v(const float* __restrict__ cf,
                                                const float* __restrict__ sc,
                                                const float* __restrict__ sh,
                                                const int* __restrict__ src,
                                                const int* __restrict__ dst,
                                                const float* __restrict__ ew,
                                                float* __restrict__ aggv, int E) {
  int e = blockIdx.x * 256 + threadIdx.x;
  if (e < E) {
    float v = (cf[src[e]] * sc[0] + sh[0]) * ew[e];
    gAddF(&aggv[dst[e]], v);
  }
}

__global__ __launch_bounds__(256) void k_agg0_c(const float* __restrict__ vf,
                                                const float* __restrict__ sc,
                                                const float* __restrict__ sh,
                                                const int* __restrict__ src,
                                                const int* __restrict__ dst,
                                                const float* __restrict__ ew,
                                                float* __restrict__ aggc, int E) {
  int e = blockIdx.x * 256 + threadIdx.x;
  if (e < E) {
    int d = dst[e], s = src[e];
    float w = ew[e];
#pragma unroll
    for (int f = 0; f < 9; ++f)
      gAddF(&aggc[(size_t)s * 9 + f], (vf[(size_t)d * 9 + f] * sc[f] + sh[f]) * w);
  }
}

// ---------------------------------------------------------------- layer-0 dense (tiny K, VALU)
__global__ __launch_bounds__(256) void k_dense0_var(const float* __restrict__ vf,
                                                    const float* __restrict__ aggv,
                                                    const float* __restrict__ invv,
                                                    const float* __restrict__ sc,
                                                    const float* __restrict__ sh,
                                                    const float* __restrict__ Wrel,   // [64]
                                                    const float* __restrict__ brel,   // [64]
                                                    const float* __restrict__ Wroot,  // [9,64]
                                                    float* __restrict__ out, int n) {
  int node = (blockIdx.x * 256 + threadIdx.x) >> 5;
  int lane = threadIdx.x & 31;
  if (node >= n) return;
  float a = aggv[node] * invv[node];
  float fb[9];
#pragma unroll
  for (int f = 0; f < 9; ++f) fb[f] = vf[(size_t)node * 9 + f] * sc[f] + sh[f];
#pragma unroll
  for (int j = 0; j < 2; ++j) {
    int h = lane + 32 * j;
    float acc = a * Wrel[h] + brel[h];
#pragma unroll
    for (int f = 0; f < 9; ++f) acc += fb[f] * Wroot[f * HD + h];
    out[(size_t)node * HD + h] = fmaxf(acc, 0.0f);
  }
}

__global__ __launch_bounds__(256) void k_dense0_cstr(const float* __restrict__ cf,
                                                     const float* __restrict__ aggc,
                                                     const float* __restrict__ invc,
                                                     const float* __restrict__ sc,
                                                     const float* __restrict__ sh,
                                                     const float* __restrict__ Wrel,   // [9,64]
                                                     const float* __restrict__ brel,   // [64]
                                                     const float* __restrict__ Wroot,  // [64]
                                                     float* __restrict__ out, int n) {
  int node = (blockIdx.x * 256 + threadIdx.x) >> 5;
  int lane = threadIdx.x & 31;
  if (node >= n) return;
  float xb = cf[node] * sc[0] + sh[0];
  float rd = invc[node];
  float ab[9];
#pragma unroll
  for (int f = 0; f < 9; ++f) ab[f] = aggc[(size_t)node * 9 + f] * rd;
#pragma unroll
  for (int j = 0; j < 2; ++j) {
    int h = lane + 32 * j;
    float acc = xb * Wroot[h] + brel[h];
#pragma unroll
    for (int f = 0; f < 9; ++f) acc += ab[f] * Wrel[f * HD + h];
    out[(size_t)node * HD + h] = fmaxf(acc, 0.0f);
  }
}

// ---------------------------------------------------------------- H=64 edge aggregation
// one wave per edge; lane covers 2 columns; BN of source fused.
__global__ __launch_bounds__(256) void k_aggregate64(const float* __restrict__ xsrc,
                                                     const float* __restrict__ sc,
                                                     const float* __restrict__ sh,
                                                     const int* __restrict__ srcIdx,
                                                     const int* __restrict__ dstIdx,
                                                     const float* __restrict__ ew,
                                                     float* __restrict__ agg, int E) {
  int e = (blockIdx.x * 256 + threadIdx.x) >> 5;
  int lane = threadIdx.x & 31;
  if (e >= E) return;
  int s = srcIdx[e], d = dstIdx[e];
  float w = ew[e];
  int c = lane * 2;
  const float* row = xsrc + (size_t)s * HD;
  float v0 = (row[c] * sc[c] + sh[c]) * w;
  float v1 = (row[c + 1] * sc[c + 1] + sh[c + 1]) * w;
  gAddF(&agg[(size_t)d * HD + c], v0);
  gAddF(&agg[(size_t)d * HD + c + 1], v1);
}

// ---------------------------------------------------------------- WMMA GEMM layer (H=64)
// out = relu( (agg * inv_deg) @ Wrel + bn(x) @ Wroot + brel ), fp32 WMMA 16x16x4.
// 8 waves/block, one 16-row tile per wave; weights staged in LDS.
__global__ __launch_bounds__(256) void k_gemm_layer(const float* __restrict__ agg,
                                                    const float* __restrict__ invdeg,
                                                    const float* __restrict__ x,
                                                    const float* __restrict__ sc,
                                                    const float* __restrict__ sh,
                                                    const float* __restrict__ Wrel,
                                                    const float* __restrict__ Wroot,
                                                    const float* __restrict__ brel,
                                                    float* __restrict__ out, int n) {
  __shared__ float sWrel[HD * HD];
  __shared__ float sWroot[HD * HD];
  __shared__ float sSc[HD], sSh[HD], sB[HD];
  int t = threadIdx.x;
  for (int i = t; i < HD * HD; i += 256) {
    sWrel[i] = Wrel[i];
    sWroot[i] = Wroot[i];
  }
  if (t < HD) {
    sSc[t] = sc[t];
    sSh[t] = sh[t];
    sB[t] = brel[t];
  }
  __syncthreads();

  int wave = t >> 5, lane = t & 31;
  int row0 = blockIdx.x * 128 + wave * 16;
  if (row0 + 16 > n) return;  // n is a multiple of 16 here
  int half = lane >> 4;       // 0: K+0/1, 1: K+2/3
  int mrow = row0 + (lane & 15);
  int ncol = lane & 15;

  float rdv = invdeg[mrow];
  const float* aggRow = agg + (size_t)mrow * HD;
  const float* xRow = x + (size_t)mrow * HD;

  // A fragments for all 16 K-steps (A: 16x4 fp32, lane<16 -> K,K+1; lane>=16 -> K+2,K+3)
  v2f Aa[16], Ax[16];
#pragma unroll
  for (int kk = 0; kk < 16; ++kk) {
    int k = kk * 4 + half * 2;
    v2f a0 = {aggRow[k] * rdv, aggRow[k + 1] * rdv};
    v2f a1 = {xRow[k] * sSc[k] + sSh[k], xRow[k + 1] * sSc[k + 1] + sSh[k + 1]};
    Aa[kk] = a0;
    Ax[kk] = a1;
  }

#pragma unroll
  for (int nt = 0; nt < 4; ++nt) {
    v8f c = {};
#pragma unroll
    for (int kk = 0; kk < 16; ++kk) {
      int k = kk * 4 + half * 2;
      v2f b = {sWrel[k * HD + nt * 16 + ncol], sWrel[(k + 1) * HD + nt * 16 + ncol]};
      c = __builtin_amdgcn_wmma_f32_16x16x4_f32(false, Aa[kk], false, b, (short)0, c,
                                                false, false);
    }
#pragma unroll
    for (int kk = 0; kk < 16; ++kk) {
      int k = kk * 4 + half * 2;
      v2f b = {sWroot[k * HD + nt * 16 + ncol], sWroot[(k + 1) * HD + nt * 16 + ncol]};
      c = __builtin_amdgcn_wmma_f32_16x16x4_f32(false, Ax[kk], false, b, (short)0, c,
                                                false, false);
    }
    float bias = sB[nt * 16 + ncol];
#pragma unroll
    for (int r = 0; r < 8; ++r) {  // C layout: VGPR r -> M = r + 8*half
      float v = c[r] + bias;
      out[(size_t)(row0 + r + 8 * half) * HD + nt * 16 + ncol] = fmaxf(v, 0.0f);
    }
  }
}

// ---------------------------------------------------------------- pooling
__global__ __launch_bounds__(256) void k_pool_sum(const float* __restrict__ x,
                                                  const int* __restrict__ batch, int n,
                                                  float* __restrict__ gsum,  // [32*64]
                                                  float* __restrict__ gcnt)  // [32]
{
  __shared__ float ls[32 * 64];
  __shared__ float lc[32];
  for (int i = threadIdx.x; i < 32 * 64; i += 256) ls[i] = 0.0f;
  if (threadIdx.x < 32) lc[threadIdx.x] = 0.0f;
  __syncthreads();
  int c = threadIdx.x & 63;
  int sub = threadIdx.x >> 6;  // 0..3
  int base = blockIdx.x * 512;
  for (int i = sub; i < 512; i += 4) {
    int node = base + i;
    if (node >= n) break;
    int b = batch[node];
    atomicAdd(&ls[b * 64 + c], x[(size_t)node * HD + c]);
    if (c == 0) atomicAdd(&lc[b], 1.0f);
  }
  __syncthreads();
  for (int i = threadIdx.x; i < 32 * 64; i += 256) {
    float v = ls[i];
    if (v != 0.0f) gAddF(&gsum[i], v);
  }
  if (threadIdx.x < 32) {
    float v = lc[threadIdx.x];
    if (v != 0.0f) gAddF(&gcnt[threadIdx.x], v);
  }
}

__global__ __launch_bounds__(256) void k_finalize_out(const float* __restrict__ sumv,
                                                      const float* __restrict__ cntv,
                                                      const float* __restrict__ sumc,
                                                      const float* __restrict__ cntc,
                                                      float* __restrict__ out) {
  int i = blockIdx.x * 256 + threadIdx.x;
  if (i < 32 * 64) {
    int b = i >> 6, h = i & 63;
    out[b * 128 + h] = sumv[i] / fmaxf(cntv[b], 1.0f);
    out[b * 128 + 64 + h] = sumc[i] / fmaxf(cntc[b], 1.0f);
  }
}

// ---------------------------------------------------------------- host
extern "C" void kernel_launch(void* const* d_in, const int* in_sizes, int n_in,
                              void* d_out, int out_size, void* d_ws, size_t ws_size,
                              hipStream_t stream) {
  const float* var_feats = (const float*)d_in[0];
  const float* cstr_feats = (const float*)d_in[1];
  const float* edge_attr = (const float*)d_in[2];
  const float* Wrel_v0 = (const float*)d_in[3];
  const float* brel_v0 = (const float*)d_in[4];
  const float* Wroot_v0 = (const float*)d_in[5];
  const float* Wrel_c0 = (const float*)d_in[6];
  const float* brel_c0 = (const float*)d_in[7];
  const float* Wroot_c0 = (const float*)d_in[8];
  const float* g_v0 = (const float*)d_in[9];
  const float* b_v0 = (const float*)d_in[10];
  const float* g_c0 = (const float*)d_in[11];
  const float* b_c0 = (const float*)d_in[12];
  const float* Wrel_v = (const float*)d_in[13];
  const float* brel_v = (const float*)d_in[14];
  const float* Wroot_v = (const float*)d_in[15];
  const float* Wrel_c = (const float*)d_in[16];
  const float* brel_c = (const float*)d_in[17];
  const float* Wroot_c = (const float*)d_in[18];
  const float* g_v = (const float*)d_in[19];
  const float* b_v = (const float*)d_in[20];
  const float* g_c = (const float*)d_in[21];
  const float* b_c = (const float*)d_in[22];
  const int* ei = (const int*)d_in[23];
  const int* var_batch = (const int*)d_in[24];
  const int* cstr_batch = (const int*)d_in[25];

  const int E = in_sizes[2];
  const int NV = in_sizes[24];
  const int NC = in_sizes[25];
  const int* src_c = ei;      // edge_index[0]
  const int* dst_v = ei + E;  // edge_index[1]

  // workspace layout (floats)
  float* W = (float*)d_ws;
  size_t oXvA = 0;
  size_t oXvB = oXvA + (size_t)NV * HD;
  size_t oXcA = oXvB + (size_t)NV * HD;
  size_t oXcB = oXcA + (size_t)NC * HD;
  size_t oAggV = oXcB + (size_t)NC * HD;   // NV*64 (layer0: NV*1)
  size_t oAggC = oAggV + (size_t)NV * HD;  // NC*64 (layer0: NC*9)
  size_t oInvV = oAggC + (size_t)NC * HD;
  size_t oInvC = oInvV + (size_t)NV;
  size_t oStats = oInvC + (size_t)NC;  // [0,128): v  [128,256): c  [256,274): v0  [274,276): c0
  size_t oScSh = oStats + 512;  // scV 0-63, shV 64-127, scC 128-191, shC 192-255,
                                // scV0 256-264, shV0 265-273, scC0 274, shC0 275
  size_t oPool = oScSh + 512;   // sumV 2048, sumC 2048, cntV 32, cntC 32

  float* xvA = W + oXvA;
  float* xvB = W + oXvB;
  float* xcA = W + oXcA;
  float* xcB = W + oXcB;
  float* aggV = W + oAggV;
  float* aggC = W + oAggC;
  float* invV = W + oInvV;
  float* invC = W + oInvC;
  float* stats = W + oStats;
  float* scsh = W + oScSh;
  float* sumV = W + oPool;
  float* sumC = sumV + 2048;
  float* cntV = sumC + 2048;
  float* cntC = cntV + 32;

  float* scV = scsh + 0;
  float* shV = scsh + 64;
  float* scC = scsh + 128;
  float* shC = scsh + 192;
  float* scV0 = scsh + 256;
  float* shV0 = scsh + 265;
  float* scC0 = scsh + 274;
  float* shC0 = scsh + 275;

  const int eb = (E + 255) / 256;

  // degrees -> reciprocal
  hipMemsetAsync(invV, 0, (size_t)NV * 4, stream);
  hipMemsetAsync(invC, 0, (size_t)NC * 4, stream);
  k_count_deg<<<eb, 256, 0, stream>>>(src_c, dst_v, E, invC, invV);
  k_inv_clamp<<<(NV + 255) / 256, 256, 0, stream>>>(invV, NV);
  k_inv_clamp<<<(NC + 255) / 256, 256, 0, stream>>>(invC, NC);

  // ---- layer 0 ----
  hipMemsetAsync(stats, 0, 512 * 4, stream);
  k_bn_stats_small<<<512, 256, 0, stream>>>(var_feats, NV * 9, 9, stats + 256);
  k_bn_stats_small<<<128, 256, 0, stream>>>(cstr_feats, NC, 1, stats + 274);
  k_bn_finalize<<<1, 64, 0, stream>>>(stats + 256, 9, 1.0f / NV, g_v0, b_v0, scV0, shV0);
  k_bn_finalize<<<1, 64, 0, stream>>>(stats + 274, 1, 1.0f / NC, g_c0, b_c0, scC0, shC0);

  hipMemsetAsync(aggV, 0, (size_t)NV * 4, stream);
  hipMemsetAsync(aggC, 0, (size_t)NC * 9 * 4, stream);
  k_agg0_v<<<eb, 256, 0, stream>>>(cstr_feats, scC0, shC0, src_c, dst_v, edge_attr, aggV, E);
  k_agg0_c<<<eb, 256, 0, stream>>>(var_feats, scV0, shV0, src_c, dst_v, edge_attr, aggC, E);

  k_dense0_var<<<(NV * 32 + 255) / 256, 256, 0, stream>>>(
      var_feats, aggV, invV, scV0, shV0, Wrel_v0, brel_v0, Wroot_v0, xvA, NV);
  k_dense0_cstr<<<(NC * 32 + 255) / 256, 256, 0, stream>>>(
      cstr_feats, aggC, invC, scC0, shC0, Wrel_c0, brel_c0, Wroot_c0, xcA, NC);

  // ---- layers 1..3 ----
  float* xv = xvA;
  float* xvn = xvB;
  float* xc = xcA;
  float* xcn = xcB;
  for (int i = 0; i < 3; ++i) {
    hipMemsetAsync(stats, 0, 256 * 4, stream);
    k_bn_stats64<<<960, 256, 0, stream>>>(xv, NV, stats);
    k_bn_stats64<<<960, 256, 0, stream>>>(xc, NC, stats + 128);
    k_bn_finalize<<<1, 64, 0, stream>>>(stats, 64, 1.0f / NV, g_v + i * 64, b_v + i * 64,
                                        scV, shV);
    k_bn_finalize<<<1, 64, 0, stream>>>(stats + 128, 64, 1.0f / NC, g_c + i * 64,
                                        b_c + i * 64, scC, shC);

    hipMemsetAsync(aggV, 0, (size_t)NV * HD * 4, stream);
    hipMemsetAsync(aggC, 0, (size_t)NC * HD * 4, stream);
    // var side: messages bn(xc)[src_c] -> aggV[dst_v]
    k_aggregate64<<<(E + 7) / 8, 256, 0, stream>>>(xc, scC, shC, src_c, dst_v, edge_attr,
                                                   aggV, E);
    // cstr side: messages bn(xv)[dst_v] -> aggC[src_c]
    k_aggregate64<<<(E + 7) / 8, 256, 0, stream>>>(xv, scV, shV, dst_v, src_c, edge_attr,
                                                   aggC, E);

    k_gemm_layer<<<(NV + 127) / 128, 256, 0, stream>>>(
        aggV, invV, xv, scV, shV, Wrel_v + (size_t)i * 4096, Wroot_v + (size_t)i * 4096,
        brel_v + i * 64, xvn, NV);
    k_gemm_layer<<<(NC + 127) / 128, 256, 0, stream>>>(
        aggC, invC, xc, scC, shC, Wrel_c + (size_t)i * 4096, Wroot_c + (size_t)i * 4096,
        brel_c + i * 64, xcn, NC);

    float* t;
    t = xv; xv = xvn; xvn = t;
    t = xc; xc = xcn; xcn = t;
  }

  // ---- pooling ----
  hipMemsetAsync(sumV, 0, 4160 * 4, stream);
  k_pool_sum<<<(NV + 511) / 512, 256, 0, stream>>>(xv, var_batch, NV, sumV, cntV);
  k_pool_sum<<<(NC + 511) / 512, 256, 0, stream>>>(xc, cstr_batch, NC, sumC, cntC);
  k_finalize_out<<<8, 256, 0, stream>>>(sumV, cntV, sumC, cntC, (float*)d_out);
}